// TransitionLayer_73770358276587
// MI455X (gfx1250) — compile-verified
//
#include <hip/hip_runtime.h>
#include <math.h>

typedef __attribute__((ext_vector_type(2))) float v2f;
typedef __attribute__((ext_vector_type(8))) float v8f;

namespace {
constexpr int kN  = 10;   // N
constexpr int kM  = 22;   // M
constexpr int kMp = 24;   // M padded to multiple of 4 (WMMA K-chunks)
constexpr int kR  = 2;    // R
constexpr int kNP = kN * kN * kR;  // 200 (i,j,r) combos
}

// grid: (ceil(B/128), 10); block: 256 threads = 8 waves; wave handles 16 batches, block row i = blockIdx.y
__global__ __launch_bounds__(256) void transition_layer_kernel(
    const float* __restrict__ pred,   // (B, 22)
    const float* __restrict__ sel,    // (10,10,2,22,3)
    const float* __restrict__ alog,   // (10,10,2)
    float* __restrict__ out,          // (B,10,10)
    int B)
{
  __shared__ float sA[kNP][kMp];   // s0' + s2'   (padded with 1.0 -> log2 = 0)
  __shared__ float sC[kNP][kMp];   // s1' - s2'   (padded with 0.0)
  __shared__ float sAl[kNP];       // sigmoid(alog)

  const int tid = threadIdx.x;

  // ---- Phase 1: wave-uniform coefficient precompute (softmax over last axis, T=0.8) ----
  for (int e = tid; e < kNP * kMp; e += 256) {
    const int p = e / kMp;
    const int m = e - p * kMp;
    float Av = 1.0f, Cv = 0.0f;
    if (m < kM) {
      const float* s3 = sel + (p * kM + m) * 3;
      const float x0 = s3[0] * 1.25f, x1 = s3[1] * 1.25f, x2 = s3[2] * 1.25f;
      const float mx3 = fmaxf(x0, fmaxf(x1, x2));
      const float e0 = __expf(x0 - mx3), e1 = __expf(x1 - mx3), e2 = __expf(x2 - mx3);
      const float inv = 1.0f / (e0 + e1 + e2);
      Av = (e0 + e2) * inv;
      Cv = (e1 - e2) * inv;
    }
    sA[p][m] = Av;
    sC[p][m] = Cv;
  }
  for (int e = tid; e < kNP; e += 256) {
    sAl[e] = 1.0f / (1.0f + __expf(-alog[e]));
  }
  __syncthreads();

  // ---- Phase 2: per-wave, 16 batch elements (one per WMMA column) ----
  const int wave = tid >> 5;
  const int lane = tid & 31;
  const int half = lane >> 4;   // which half-wave (K-slot group)
  const int col  = lane & 15;   // batch column within wave tile
  const int i    = blockIdx.y;  // output row (causal: j >= i active)

  const int b     = blockIdx.x * 128 + wave * 16 + col;
  const bool valid = (b < B);
  const int bl    = valid ? b : (B - 1);

  // Each lane needs u[b, m] for m = 4t + 2*half + {0,1}, t = 0..5 (12 values)
  float uu[12];
  const float* ub = pred + (long long)bl * kM;
#pragma unroll
  for (int t = 0; t < 6; ++t) {
    const int m0 = 4 * t + 2 * half;
    uu[2 * t + 0] = (m0     < kM) ? ub[m0]     : 0.0f;
    uu[2 * t + 1] = (m0 + 1 < kM) ? ub[m0 + 1] : 0.0f;
  }

  // A-operand = all ones: D[row,n] = sum_k B[k,n] (column sums; layout-invariant)
  v2f ones; ones[0] = 1.0f; ones[1] = 1.0f;

  float lg[kN];
  float mx = -3.0e38f;
#pragma unroll
  for (int j = 0; j < kN; ++j) {
    float clause[kR];
#pragma unroll
    for (int r = 0; r < kR; ++r) {
      const int p = (i * kN + j) * kR + r;
      v8f acc = {0.f, 0.f, 0.f, 0.f, 0.f, 0.f, 0.f, 0.f};
#pragma unroll
      for (int t = 0; t < 6; ++t) {
        const int m0 = 4 * t + 2 * half;
        // lit in (0,1]; padded slots give lit = 1 -> log2 = 0 exactly
        const float lit0 = sA[p][m0]     + sC[p][m0]     * uu[2 * t + 0];
        const float lit1 = sA[p][m0 + 1] + sC[p][m0 + 1] * uu[2 * t + 1];
        v2f bb;
        bb[0] = __log2f(lit0);
        bb[1] = __log2f(lit1);
        // D = ones(16x4) x B(4x16) + C : every lane's acc[0] = sum_m log2(lit) for its batch column
        acc = __builtin_amdgcn_wmma_f32_16x16x4_f32(
            false, ones, false, bb, (short)0, acc, false, false);
      }
      clause[r] = __builtin_amdgcn_exp2f(acc[0]);
    }
    const int ij = i * kN + j;
    const float g = 1.0f - (1.0f - sAl[ij * 2 + 0] * clause[0])
                         * (1.0f - sAl[ij * 2 + 1] * clause[1]);
    const float l = __logf(fmaxf(g, 1e-6f));
    lg[j] = (j >= i) ? l : -3.0e38f;   // mask: -inf for j < i
    mx = fmaxf(mx, lg[j]);
  }

  // Row softmax with T = 0.5 (x2); j < i contributes exactly 0
  float ex[kN];
  float sum = 0.0f;
#pragma unroll
  for (int j = 0; j < kN; ++j) {
    ex[j] = (j >= i) ? __expf(2.0f * (lg[j] - mx)) : 0.0f;
    sum += ex[j];
  }
  const float inv = 1.0f / sum;

  if (valid && half == 0) {  // halves are duplicates; one writer per batch row
    float* op = out + ((long long)b * kN + i) * kN;
#pragma unroll
    for (int j = 0; j < kN; ++j) op[j] = ex[j] * inv;
  }
}

extern "C" void kernel_launch(void* const* d_in, const int* in_sizes, int n_in,
                              void* d_out, int out_size, void* d_ws, size_t ws_size,
                              hipStream_t stream) {
  const float* pred = (const float*)d_in[0];
  const float* sel  = (const float*)d_in[1];
  const float* alog = (const float*)d_in[2];
  // d_in[3] (mask) is statically known upper-triangular; handled analytically.
  float* out = (float*)d_out;

  const int B = in_sizes[0] / kM;          // 16384
  dim3 grid((B + 127) / 128, kN, 1);       // (128, 10)
  transition_layer_kernel<<<grid, 256, 0, stream>>>(pred, sel, alog, out, B);
}